// SSIM_33157147525351
// MI455X (gfx1250) — compile-verified
//
#include <hip/hip_runtime.h>

// SSIM on MI455X (gfx1250, wave32).
// Separable 11-tap Gaussian conv cast as banded matmuls on V_WMMA_F32_16X16X4_F32
// (fp32 matrix pipe: keeps sigma = E[x^2]-mu^2 cancellation-safe, ~8x the fp32
// VALU rate, moving the kernel from VALU-compute-bound (~35us) toward the
// 8.6us HBM floor for the 200MB of compulsory input traffic).

typedef float v2f __attribute__((ext_vector_type(2)));
typedef float v8f __attribute__((ext_vector_type(8)));

#define IMG_H 512
#define IMG_W 512
#define OUT_H 502
#define OUT_W 502
#define PLANES 96            // B*C = 32*3
#define WAVES_PER_BLOCK 4
#define XSTR 33              // LDS row stride for 28-row input tiles
#define VSTR 33              // LDS row stride for 16x32 intermediate

// Gaussian taps, sigma=1.5, size 11, padded with zeros so a clamped index
// outside the band reads 0.0f (banded-matrix fill).
__constant__ float GW16[16] = {
    0.00102838f, 0.00759874f, 0.03600077f, 0.10936069f, 0.21300553f,
    0.26601172f,
    0.21300553f, 0.10936069f, 0.03600077f, 0.00759874f, 0.00102838f,
    0.0f, 0.0f, 0.0f, 0.0f, 0.0f};

__device__ __forceinline__ float gband(int i) {
  unsigned u = (unsigned)i;
  if (u > 15u) u = 15u;      // negative or >10 -> zero slot
  return GW16[u];
}

__global__ __launch_bounds__(128) void ssim_wmma_kernel(
    const float* __restrict__ img1, const float* __restrict__ img2,
    float* __restrict__ out) {
  __shared__ float sX[WAVES_PER_BLOCK][28 * XSTR];
  __shared__ float sY[WAVES_PER_BLOCK][28 * XSTR];
  __shared__ float sV[WAVES_PER_BLOCK][16 * VSTR];

  const int wave = threadIdx.x >> 5;
  const int lane = threadIdx.x & 31;
  const int hf   = lane >> 4;   // lane half: selects K pair / M+8 rows
  const int l16  = lane & 15;

  // One wave == one 16x16 output tile. 96 planes * 32*32 tiles.
  const int tile  = blockIdx.x * WAVES_PER_BLOCK + wave;
  const int plane = tile >> 10;
  const int trem  = tile & 1023;
  const int h0 = (trem >> 5) << 4;
  const int w0 = (trem & 31) << 4;

  const size_t pbase = (size_t)plane * (IMG_H * IMG_W);
  float* Xs = sX[wave];
  float* Ys = sY[wave];
  float* Vs = sV[wave];

  // Stage the 28x32 input window for both images (zero padded at the image
  // edge so all lanes stay active: WMMA requires EXEC == all ones).
  #pragma unroll 4
  for (int r = 0; r < 28; ++r) {
    const int h = h0 + r;
    const int w = w0 + lane;
    const bool ok = (h < IMG_H) && (w < IMG_W);
    const size_t g = pbase + (size_t)h * IMG_W + (size_t)w;
    Xs[r * XSTR + lane] = ok ? img1[g] : 0.0f;
    Ys[r * XSTR + lane] = ok ? img2[g] : 0.0f;
  }

  // Banded Gaussian operand. Identical per-lane layout for the vertical
  // A-matrix (A[m,k] = g[k-m]) and horizontal B-matrix (B[k,n] = g[k-n]):
  // element(vgpr v) = g[(kk*4 + 2*hf + v) - (lane&15)].
  v2f band[7];
  #pragma unroll
  for (int kk = 0; kk < 7; ++kk) {
    const int k = kk * 4 + hf * 2;
    band[kk].x = gband(k - l16);
    band[kk].y = gband(k + 1 - l16);
  }

  float res[5][8];

  #pragma unroll
  for (int map = 0; map < 5; ++map) {
    // ---- vertical pass: V(16x32) = Gband(16x28) x E(28x32) ----
    #pragma unroll
    for (int cg = 0; cg < 2; ++cg) {
      v8f acc = {};
      const int col = cg * 16 + l16;
      #pragma unroll
      for (int kk = 0; kk < 7; ++kk) {
        const int k  = kk * 4 + hf * 2;
        const int i0 = k * XSTR + col;
        const float x0 = Xs[i0],        y0 = Ys[i0];
        const float x1 = Xs[i0 + XSTR], y1 = Ys[i0 + XSTR];
        v2f b;
        b.x = (map == 0) ? x0 : (map == 1) ? y0 : (map == 2) ? x0 * x0
                                           : (map == 3) ? y0 * y0 : x0 * y0;
        b.y = (map == 0) ? x1 : (map == 1) ? y1 : (map == 2) ? x1 * x1
                                           : (map == 3) ? y1 * y1 : x1 * y1;
        acc = __builtin_amdgcn_wmma_f32_16x16x4_f32(
            false, band[kk], false, b, (short)0, acc, false, false);
      }
      // D layout: VGPR r -> row r (lanes 0-15) / row r+8 (lanes 16-31).
      #pragma unroll
      for (int r = 0; r < 8; ++r)
        Vs[(r + 8 * hf) * VSTR + col] = acc[r];
    }

    // ---- horizontal pass: Out(16x16) = V(16x28) x Gband(28x16) ----
    v8f acc = {};
    #pragma unroll
    for (int kk = 0; kk < 7; ++kk) {
      const int k = kk * 4 + hf * 2;
      v2f a;
      a.x = Vs[l16 * VSTR + k];
      a.y = Vs[l16 * VSTR + k + 1];
      acc = __builtin_amdgcn_wmma_f32_16x16x4_f32(
          false, a, false, band[kk], (short)0, acc, false, false);
    }
    #pragma unroll
    for (int r = 0; r < 8; ++r) res[map][r] = acc[r];
  }

  // ---- SSIM combine + mean reduction ----
  const float C1v = 1.0e-4f;   // 0.01^2
  const float C2v = 9.0e-4f;   // 0.03^2
  float sum = 0.0f;
  #pragma unroll
  for (int r = 0; r < 8; ++r) {
    const int oh = h0 + r + 8 * hf;
    const int ow = w0 + l16;
    if (oh < OUT_H && ow < OUT_W) {
      const float mu1 = res[0][r], mu2 = res[1][r];
      const float e11 = res[2][r], e22 = res[3][r], e12 = res[4][r];
      const float mu1s = mu1 * mu1, mu2s = mu2 * mu2, m12 = mu1 * mu2;
      const float s1 = e11 - mu1s, s2 = e22 - mu2s, s12 = e12 - m12;
      const float num = (2.0f * m12 + C1v) * (2.0f * s12 + C2v);
      const float den = (mu1s + mu2s + C1v) * (s1 + s2 + C2v);
      sum += num / den;
    }
  }
  #pragma unroll
  for (int off = 16; off > 0; off >>= 1)
    sum += __shfl_xor(sum, off, 32);
  if (lane == 0)
    atomicAdd(out, sum * (1.0f / ((float)PLANES * (float)OUT_H * (float)OUT_W)));
}

extern "C" void kernel_launch(void* const* d_in, const int* in_sizes, int n_in,
                              void* d_out, int out_size, void* d_ws,
                              size_t ws_size, hipStream_t stream) {
  (void)in_sizes; (void)n_in; (void)out_size; (void)d_ws; (void)ws_size;
  const float* img1 = (const float*)d_in[0];
  const float* img2 = (const float*)d_in[1];
  float* out = (float*)d_out;

  hipMemsetAsync(out, 0, sizeof(float), stream);  // graph-capture safe
  const int total_tiles = PLANES * 32 * 32;       // 98304
  const int blocks = total_tiles / WAVES_PER_BLOCK;
  ssim_wmma_kernel<<<blocks, 32 * WAVES_PER_BLOCK, 0, stream>>>(img1, img2, out);
}